// GraphConvEncoder_74861279969731
// MI455X (gfx1250) — compile-verified
//
#include <hip/hip_runtime.h>
#include <hip/hip_bf16.h>

#define N_FEAT 40          // D
#define KCHUNKS 10         // 40 / 4
#define CTILES 3           // ceil(40/16)
#define NPAD 48            // padded column count (3 * 16)

typedef __attribute__((ext_vector_type(2))) float v2f;
typedef __attribute__((ext_vector_type(8))) float v8f;

// Guaranteed hardware f32 atomic add (no-return form, device scope).
// Verbatim CDNA5 instruction; tracked with STOREcnt, waited by s_endpgm.
__device__ __forceinline__ void gatomic_add_f32(float* p, float v) {
    asm volatile("global_atomic_add_f32 %0, %1, off scope:SCOPE_DEV"
                 :: "v"(p), "v"(v) : "memory");
}

// ---------------------------------------------------------------------------
// Zero the two aggregation buffers (8M floats) with float4 stores.
// ---------------------------------------------------------------------------
__global__ void gc_zero_kernel(float4* __restrict__ p, long long n4) {
    long long i = (long long)blockIdx.x * blockDim.x + threadIdx.x;
    if (i < n4) p[i] = make_float4(0.f, 0.f, 0.f, 0.f);
}

// ---------------------------------------------------------------------------
// Edge scatter-add:  agg[dst] += feat[src]   (segment_sum over edges)
// 10 threads per edge, each owns one float4 chunk of the 40-float row.
// ---------------------------------------------------------------------------
__global__ void gc_scatter_kernel(const int* __restrict__ eidx,   // [2, E]
                                  const float* __restrict__ feat, // [N, 40]
                                  float* __restrict__ agg,        // [N, 40]
                                  int n_edges) {
    long long tid = (long long)blockIdx.x * blockDim.x + threadIdx.x;
    long long total = (long long)n_edges * KCHUNKS;
    if (tid >= total) return;
    int e = (int)(tid / KCHUNKS);
    int c = (int)(tid % KCHUNKS);
    int src = eidx[e];
    int dst = eidx[n_edges + e];
    const float4 v = *(const float4*)(feat + (size_t)src * N_FEAT + c * 4);
    float* o = agg + (size_t)dst * N_FEAT + c * 4;
    gatomic_add_f32(o + 0, v.x);
    gatomic_add_f32(o + 1, v.y);
    gatomic_add_f32(o + 2, v.z);
    gatomic_add_f32(o + 3, v.w);
}

// ---------------------------------------------------------------------------
// Fused dual-GEMM + bias (+ optional ReLU), f32 WMMA 16x16x4:
//   Out[16-row tile] = Aagg @ Wrel + Ax @ Wroot + bias
// One wave per 16-row node tile; 3 column tiles of 16.
// Weights staged zero-padded [40][48] in LDS once per block, so the inner
// loop B-fragment fetches are unguarded ds_load_b32 (pad columns = 0).
// Layouts per CDNA5 ISA 7.12.2:
//   A (16x4 f32): lane L holds row (L&15); VGPR0/1 = K {0,1} for lanes 0-15,
//                 K {2,3} for lanes 16-31.
//   B (4x16 f32): symmetric — lane L holds column N=(L&15), same K packing.
//   C/D (16x16 f32, 8 VGPRs): N = lane&15; VGPR r holds row r (lanes 0-15)
//                 or row r+8 (lanes 16-31).
// WMMA only executed under wave-uniform control flow (EXEC all ones).
// ---------------------------------------------------------------------------
__global__ void gc_gemm_kernel(const float* __restrict__ Aagg,  // [N, 40]
                               const float* __restrict__ Ax,    // [N, 40]
                               const float* __restrict__ Wrel,  // [40, 40]
                               const float* __restrict__ Wroot, // [40, 40]
                               const float* __restrict__ bias,  // [40]
                               float* __restrict__ Out,         // [N, 40]
                               int n_nodes, int do_relu) {
    __shared__ float sWrel [N_FEAT * NPAD];   // [40][48], cols 40..47 = 0
    __shared__ float sWroot[N_FEAT * NPAD];

    for (int i = threadIdx.x; i < N_FEAT * NPAD; i += blockDim.x) {
        const int r = i / NPAD, c = i % NPAD;
        const bool in = (c < N_FEAT);
        sWrel [i] = in ? Wrel [r * N_FEAT + c] : 0.0f;
        sWroot[i] = in ? Wroot[r * N_FEAT + c] : 0.0f;
    }
    __syncthreads();

    const int wave = threadIdx.x >> 5;
    const int lane = threadIdx.x & 31;
    const int tile = blockIdx.x * (blockDim.x >> 5) + wave;
    const int row0 = tile * 16;
    if (row0 >= n_nodes) return;   // wave-uniform exit: EXEC stays all-1s

    const int mrow = lane & 15;              // A row in tile / B column in ctile
    const int kb   = (lane < 16) ? 0 : 2;    // K sub-offset for this lane

    v8f acc[CTILES];
    for (int ct = 0; ct < CTILES; ++ct) acc[ct] = (v8f)(0.0f);

    const float* arow_agg = Aagg + (size_t)(row0 + mrow) * N_FEAT;
    const float* arow_x   = Ax   + (size_t)(row0 + mrow) * N_FEAT;

    for (int kc = 0; kc < KCHUNKS; ++kc) {
        const int k0 = kc * 4 + kb;          // even -> aligned float2 loads
        const float2 ag = *(const float2*)(arow_agg + k0);
        const float2 ax = *(const float2*)(arow_x   + k0);
        v2f a_agg; a_agg.x = ag.x; a_agg.y = ag.y;
        v2f a_x;   a_x.x   = ax.x; a_x.y   = ax.y;

        const float* r0 = sWrel  + k0 * NPAD + mrow;   // rows k0, k0+1
        const float* q0 = sWroot + k0 * NPAD + mrow;

        #pragma unroll
        for (int ct = 0; ct < CTILES; ++ct) {
            v2f b_rel, b_root;
            b_rel.x  = r0[ct * 16];          b_rel.y  = r0[NPAD + ct * 16];
            b_root.x = q0[ct * 16];          b_root.y = q0[NPAD + ct * 16];
            acc[ct] = __builtin_amdgcn_wmma_f32_16x16x4_f32(
                false, a_agg, false, b_rel, (short)0, acc[ct], false, false);
            acc[ct] = __builtin_amdgcn_wmma_f32_16x16x4_f32(
                false, a_x, false, b_root, (short)0, acc[ct], false, false);
        }
    }

    const int rbase = row0 + ((lane < 16) ? 0 : 8);
    for (int ct = 0; ct < CTILES; ++ct) {
        const int n = ct * 16 + mrow;
        if (n >= N_FEAT) continue;
        const float bv = bias[n];
        #pragma unroll
        for (int r = 0; r < 8; ++r) {
            const int row = rbase + r;
            float v = acc[ct][r] + bv;
            if (do_relu) v = fmaxf(v, 0.0f);
            if (row < n_nodes) Out[(size_t)row * N_FEAT + n] = v;
        }
    }
}

// ---------------------------------------------------------------------------
// Host-side launcher
// ---------------------------------------------------------------------------
extern "C" void kernel_launch(void* const* d_in, const int* in_sizes, int n_in,
                              void* d_out, int out_size, void* d_ws, size_t ws_size,
                              hipStream_t stream) {
    const float* x       = (const float*)d_in[0];
    const int*   eidx    = (const int*)  d_in[1];
    const float* W1_rel  = (const float*)d_in[2];
    const float* b1      = (const float*)d_in[3];
    const float* W1_root = (const float*)d_in[4];
    const float* W2_rel  = (const float*)d_in[5];
    const float* b2      = (const float*)d_in[6];
    const float* W2_root = (const float*)d_in[7];
    float* out = (float*)d_out;

    const int n_nodes = in_sizes[0] / N_FEAT;   // 100000
    const int n_edges = in_sizes[1] / 2;        // 1600000

    float* agg1 = (float*)d_ws;                           // [N,40]
    float* agg2 = agg1 + (size_t)n_nodes * N_FEAT;        // [N,40]
    float* h    = agg2 + (size_t)n_nodes * N_FEAT;        // [N,40]

    // 1) zero agg1 + agg2 (contiguous)
    {
        long long n4 = (long long)n_nodes * N_FEAT * 2 / 4;
        int blocks = (int)((n4 + 255) / 256);
        gc_zero_kernel<<<blocks, 256, 0, stream>>>((float4*)agg1, n4);
    }

    // 2) agg1 = segment_sum(x[src], dst)
    {
        long long total = (long long)n_edges * KCHUNKS;
        int blocks = (int)((total + 255) / 256);
        gc_scatter_kernel<<<blocks, 256, 0, stream>>>(eidx, x, agg1, n_edges);
    }

    // 3) h = relu(agg1 @ W1_rel + x @ W1_root + b1)
    {
        int tiles = (n_nodes + 15) / 16;       // 6250
        int waves_per_block = 8;               // 256 threads
        int blocks = (tiles + waves_per_block - 1) / waves_per_block;
        gc_gemm_kernel<<<blocks, 256, 0, stream>>>(agg1, x, W1_rel, W1_root, b1,
                                                   h, n_nodes, 1);
    }

    // 4) agg2 = segment_sum(h[src], dst)
    {
        long long total = (long long)n_edges * KCHUNKS;
        int blocks = (int)((total + 255) / 256);
        gc_scatter_kernel<<<blocks, 256, 0, stream>>>(eidx, h, agg2, n_edges);
    }

    // 5) out = agg2 @ W2_rel + h @ W2_root + b2
    {
        int tiles = (n_nodes + 15) / 16;
        int waves_per_block = 8;
        int blocks = (tiles + waves_per_block - 1) / waves_per_block;
        gc_gemm_kernel<<<blocks, 256, 0, stream>>>(agg2, h, W2_rel, W2_root, b2,
                                                   out, n_nodes, 0);
    }
}